// ADAMEMNET_83288005804283
// MI455X (gfx1250) — compile-verified
//
#include <hip/hip_runtime.h>
#include <hip/hip_bf16.h>

typedef __attribute__((ext_vector_type(2))) float v2f;
typedef __attribute__((ext_vector_type(8))) float v8f;

#define T_STEPS 200
#define BATCH   32
#define N_IN    10000
#define N_OUT   3
#define M_TOTAL (T_STEPS * BATCH)        // 6400 rows
#define M_TILES (M_TOTAL / 16)           // 400
#define KCHUNKS 25
#define KCHUNK  (N_IN / KCHUNKS)         // 400
#define KSTEPS  (KCHUNK / 4)             // 100 WMMA f32 16x16x4 steps
#define WAVES_PER_BLOCK 4

// ---------------------------------------------------------------------------
// Kernel 1: initialize proj[t,b,o] = bias[o]  (GEMM accumulates on top)
// ---------------------------------------------------------------------------
__global__ void adamem_init(const float* __restrict__ bias,
                            float* __restrict__ proj, int total) {
    int i = blockIdx.x * blockDim.x + threadIdx.x;
    if (i < total) proj[i] = bias[i % N_OUT];
}

// ---------------------------------------------------------------------------
// Kernel 2: proj += spikes @ W^T via V_WMMA_F32_16X16X4_F32.
// One wave = one (m-tile of 16 rows, K-chunk of 400). 10000 waves total.
// A fragment (16x4 f32): lane L holds row (L&15); VGPR0 = K {0|2}, VGPR1 = K {1|3}
//   selected by lane half -> one float2 load per step per lane.
// B fragment (4x16): lane L holds column n = L&15. Only columns n<3 are ever
//   stored, and WMMA columns are independent, so lanes n>=3 simply load a
//   *valid* (clamped) W row instead of zeros -> branch-free inner loop.
// ---------------------------------------------------------------------------
__global__ void __launch_bounds__(32 * WAVES_PER_BLOCK)
adamem_gemm(const float* __restrict__ spikes,
            const float* __restrict__ W,
            float* __restrict__ proj) {
    const int lane   = threadIdx.x & 31;
    const int wid    = blockIdx.x * WAVES_PER_BLOCK + (threadIdx.x >> 5);
    const int kchunk = wid % KCHUNKS;
    const int mtile  = wid / KCHUNKS;          // 0..M_TILES-1
    const int half   = lane >> 4;              // 0: K{0,1}, 1: K{2,3}
    const int n      = lane & 15;              // output column (valid if < 3)
    const int row    = mtile * 16 + n;         // A row for this lane
    const int kbase  = kchunk * KCHUNK + half * 2;
    const int nc     = (n < N_OUT) ? n : 0;    // clamp: garbage cols duplicate col 0

    // both bases are 8-byte aligned: row*N_IN and nc*N_IN are even, kbase even
    const v2f* ap = reinterpret_cast<const v2f*>(spikes + (size_t)row * N_IN + kbase);
    const v2f* bp = reinterpret_cast<const v2f*>(W      + (size_t)nc  * N_IN + kbase);

    v8f acc = {0.f, 0.f, 0.f, 0.f, 0.f, 0.f, 0.f, 0.f};

#pragma unroll 10
    for (int s = 0; s < KSTEPS; ++s) {
        v2f a  = ap[s * 2];                    // spikes[row][kbase+4s .. +1]
        v2f bb = bp[s * 2];                    // W[nc][kbase+4s .. +1]
        acc = __builtin_amdgcn_wmma_f32_16x16x4_f32(
            /*neg_a=*/false, a, /*neg_b=*/false, bb,
            /*c_mod=*/(short)0, acc, /*reuse_a=*/false, /*reuse_b=*/false);
    }

    // C/D layout: lane half h, VGPR v -> M = v + 8*h, N = lane&15
    if (n < N_OUT) {
#pragma unroll
        for (int v = 0; v < 8; ++v) {
            int grow = mtile * 16 + v + half * 8;
            atomicAdd(&proj[grow * N_OUT + n], acc[v]);
        }
    }
}

// ---------------------------------------------------------------------------
// Kernel 3: sequential adaptive-LIF scan. 96 independent (b,o) state chains,
// 200 steps each; proj is L2-resident.
// ---------------------------------------------------------------------------
__global__ void adamem_scan(const float* __restrict__ proj,
                            float* __restrict__ out) {
    const int j = threadIdx.x;
    if (j >= BATCH * N_OUT) return;
    float* spk_rec = out;                               // [T,B,3]
    float* mem_rec = out + (size_t)T_STEPS * BATCH * N_OUT;

    float mem = 0.0f;
    float thr = 1.0f;                                   // THR_INIT
    for (int t = 0; t < T_STEPS; ++t) {
        float x = proj[t * (BATCH * N_OUT) + j];
        mem = 0.99f * mem + x;                          // BETA
        float spk = (mem > thr) ? 1.0f : 0.0f;
        mem -= spk * thr;                               // reset by subtraction
        thr = 0.95f * thr + 5.0f * spk;                 // BASE_THR=0
        spk_rec[t * (BATCH * N_OUT) + j] = spk;
        mem_rec[t * (BATCH * N_OUT) + j] = mem;
    }
}

// ---------------------------------------------------------------------------
extern "C" void kernel_launch(void* const* d_in, const int* in_sizes, int n_in,
                              void* d_out, int out_size, void* d_ws, size_t ws_size,
                              hipStream_t stream) {
    const float* spikes = (const float*)d_in[0];   // [T, B, N_IN]
    const float* W      = (const float*)d_in[1];   // [N_OUT, N_IN]
    const float* bias   = (const float*)d_in[2];   // [N_OUT]
    float* out  = (float*)d_out;                   // spk_rec ++ mem_rec
    float* proj = (float*)d_ws;                    // [T*B, N_OUT] = 76.8 KB

    const int total = T_STEPS * BATCH * N_OUT;     // 19200
    adamem_init<<<(total + 255) / 256, 256, 0, stream>>>(bias, proj, total);

    const int nblocks = (M_TILES * KCHUNKS) / WAVES_PER_BLOCK;  // 2500
    adamem_gemm<<<nblocks, 32 * WAVES_PER_BLOCK, 0, stream>>>(spikes, W, proj);

    adamem_scan<<<1, 96, 0, stream>>>(proj, out);
}